// GraphAttentionLayer_41618233098759
// MI455X (gfx1250) — compile-verified
//
#include <hip/hip_runtime.h>

// ---------------------------------------------------------------------------
// GAT layer for MI455X (gfx1250, wave32) — single-pass-over-adj version.
//   K1: h = x@W                          : v_wmma_f32_16x16x4_f32
//   K2: att_l/att_r = h @ a_{l,r}        : wave shuffle reductions
//   K3: out = elu(softmax(mask(e)) @ h)  : FUSED online-softmax + WMMA,
//       adj streamed HBM->LDS exactly once via async global->LDS loads
//       (double buffered, ASYNCcnt), non-temporal so the 256MB stream does
//       not evict h (8MB, re-read from L2) from the 192MB L2.
// HBM traffic ~ 1 pass over adj (256 MB) => ~11 us floor at 23.3 TB/s.
// ---------------------------------------------------------------------------

typedef __attribute__((ext_vector_type(2))) float v2f;   // f32 A/B frag (16x4 / 4x16)
typedef __attribute__((ext_vector_type(8))) float v8f;   // f32 C/D frag (16x16)

#define N_NODES 8192
#define IN_F    512
#define OUT_F   256
#define ALPHA   0.2f
#define KC      128                 // adj K-chunk (columns) per stage
#define NCHUNK  (N_NODES / KC)      // 64
#define LDA     132                 // padded LDS row stride (floats)

// ---- CDNA5 async global->LDS copy (ASYNCcnt), per-lane 16B, non-temporal --
__device__ __forceinline__ void async_copy_b128_nt(void* lds_dst, const void* gsrc) {
  const unsigned lds_off          = (unsigned)(unsigned long long)lds_dst;  // LDS byte offset
  const unsigned long long gaddr  = (unsigned long long)gsrc;
  asm volatile("global_load_async_to_lds_b128 %0, %1, off th:TH_LOAD_NT"
               :: "v"(lds_off), "v"(gaddr) : "memory");
}
__device__ __forceinline__ void wait_async_le2() {
  asm volatile("s_wait_asynccnt 0x2" ::: "memory");
}
__device__ __forceinline__ void wait_async_le0() {
  asm volatile("s_wait_asynccnt 0x0" ::: "memory");
}

// ---------------- K1: h = x @ W, one wave per 16x16 tile --------------------
__global__ __launch_bounds__(256) void gat_xw_kernel(const float* __restrict__ x,
                                                     const float* __restrict__ W,
                                                     float* __restrict__ h) {
  const int wave  = threadIdx.x >> 5;
  const int lane  = threadIdx.x & 31;
  const int id    = blockIdx.x * 8 + wave;   // 8192 tile-waves total
  const int m0    = (id >> 4) * 16;
  const int n0    = (id & 15) * 16;
  const int mrow  = lane & 15;               // A: M ; B/C/D: N
  const int khalf = lane >> 4;               // K pair select

  const float* xrow = x + (m0 + mrow) * IN_F + 2 * khalf;
  v8f c = {};

  #pragma unroll 4
  for (int k = 0; k < IN_F; k += 4) {
    v2f a = *(const v2f*)(xrow + k);
    v2f b;
    b.x = W[(k + 2 * khalf) * OUT_F + n0 + mrow];
    b.y = W[(k + 2 * khalf + 1) * OUT_F + n0 + mrow];
    c = __builtin_amdgcn_wmma_f32_16x16x4_f32(false, a, false, b,
                                              (short)0, c, false, false);
  }
  #pragma unroll
  for (int r = 0; r < 8; ++r)
    h[(m0 + r + 8 * khalf) * OUT_F + n0 + mrow] = c[r];
}

// ---------------- K2: att_l[i] = h[i]·a_left, att_r[i] = h[i]·a_right ------
__global__ __launch_bounds__(256) void gat_att_kernel(const float* __restrict__ h,
                                                      const float* __restrict__ a,
                                                      float* __restrict__ att_l,
                                                      float* __restrict__ att_r) {
  const int wave = threadIdx.x >> 5;
  const int lane = threadIdx.x & 31;
  const int i    = blockIdx.x * 8 + wave;
  float sl = 0.f, sr = 0.f;
  #pragma unroll
  for (int n = lane; n < OUT_F; n += 32) {
    const float hv = h[i * OUT_F + n];
    sl += hv * a[n];
    sr += hv * a[OUT_F + n];
  }
  #pragma unroll
  for (int off = 16; off > 0; off >>= 1) {
    sl += __shfl_xor(sl, off, 32);
    sr += __shfl_xor(sr, off, 32);
  }
  if (lane == 0) { att_l[i] = sl; att_r[i] = sr; }
}

// ---------------- K3: fused online-softmax aggregation ----------------------
// Block = 16 output rows x 256 cols (8 waves x two 16-wide WMMA tiles).
// adj[m0:m0+16][:] streamed once through double-buffered LDS via async copies.
__global__ __launch_bounds__(256) void gat_fused_kernel(
    const float* __restrict__ adj, const float* __restrict__ h,
    const float* __restrict__ att_l, const float* __restrict__ att_r,
    float* __restrict__ out) {
  __shared__ float ar_lds[N_NODES];           // 32 KB, shared by all chunks
  __shared__ float adj_lds[2][16 * LDA];      // 2 x 8.25 KB double buffer

  const int tid   = threadIdx.x;
  const int wave  = tid >> 5;
  const int lane  = tid & 31;
  const int m0    = blockIdx.x * 16;
  const int n0    = wave * 32;                // two 16-col tiles per wave
  const int mrow  = lane & 15;
  const int khalf = lane >> 4;

  // Preload att_r once (regular loads + DS stores; covered by first barrier).
  for (int idx = tid; idx < N_NODES / 4; idx += 256)
    ((float4*)ar_lds)[idx] = ((const float4*)att_r)[idx];

  const float al = att_l[m0 + mrow];

  float m_run = -__builtin_inff();
  float s_run = 0.f;
  v8f c0 = {}, c1 = {};

  // Issue async stage of chunk 0: wave w copies adj rows 2w, 2w+1 (512B each).
  #pragma unroll
  for (int j = 0; j < 2; ++j) {
    const int r = 2 * wave + j;
    async_copy_b128_nt(&adj_lds[0][r * LDA + lane * 4],
                       adj + (size_t)(m0 + r) * N_NODES + lane * 4);
  }

  for (int t = 0; t < NCHUNK; ++t) {
    const int buf = t & 1;
    const int kb  = t * KC;

    if (t + 1 < NCHUNK) {               // stage next chunk into other buffer
      #pragma unroll
      for (int j = 0; j < 2; ++j) {
        const int r = 2 * wave + j;
        async_copy_b128_nt(&adj_lds[buf ^ 1][r * LDA + lane * 4],
                           adj + (size_t)(m0 + r) * N_NODES + (size_t)(kb + KC) + lane * 4);
      }
      wait_async_le2();                 // chunk t's 2 copies complete (in-order)
    } else {
      wait_async_le0();
    }
    __syncthreads();                    // all waves' copies visible in LDS

    const float* bp = &adj_lds[buf][mrow * LDA];

    // ---- pass 1: chunk max per row (this lane covers half the columns) ----
    float mc = -__builtin_inff();
    #pragma unroll 8
    for (int k = 0; k < KC; k += 4) {
      const int kk  = k + 2 * khalf;
      const float a0 = bp[kk], a1 = bp[kk + 1];
      const float v0 = al + ar_lds[kb + kk];
      const float v1 = al + ar_lds[kb + kk + 1];
      const float e0 = v0 > 0.f ? v0 : ALPHA * v0;
      const float e1 = v1 > 0.f ? v1 : ALPHA * v1;
      mc = fmaxf(mc, (a0 != 0.f) ? e0 : -__builtin_inff());
      mc = fmaxf(mc, (a1 != 0.f) ? e1 : -__builtin_inff());
    }
    mc = fmaxf(mc, __shfl_xor(mc, 16, 32));            // combine half-lanes
    const float m_new = fmaxf(m_run, mc);
    const float f     = (m_run == m_new) ? 1.f : __expf(m_run - m_new);
    s_run *= f;
    m_run  = m_new;

    // rescale accumulators: factor for C-row q lives in lane q (ds_bpermute)
    #pragma unroll
    for (int r = 0; r < 8; ++r) {
      const float fr = __shfl(f, r + 8 * khalf, 32);
      c0[r] *= fr;
      c1[r] *= fr;
    }

    // ---- pass 2: P = exp(e - m_run) masked; accumulate P@h and row sum ----
    float sp = 0.f;
    #pragma unroll 8
    for (int k = 0; k < KC; k += 4) {
      const int kk  = k + 2 * khalf;
      const float a0 = bp[kk], a1 = bp[kk + 1];
      const float v0 = al + ar_lds[kb + kk];
      const float v1 = al + ar_lds[kb + kk + 1];
      const float e0 = v0 > 0.f ? v0 : ALPHA * v0;
      const float e1 = v1 > 0.f ? v1 : ALPHA * v1;
      v2f pa;
      pa.x = (a0 != 0.f) ? __expf(e0 - m_run) : 0.f;
      pa.y = (a1 != 0.f) ? __expf(e1 - m_run) : 0.f;
      sp += pa.x + pa.y;

      const int kg = kb + kk;
      v2f b0v, b1v;
      b0v.x = h[kg * OUT_F + n0 + mrow];
      b0v.y = h[(kg + 1) * OUT_F + n0 + mrow];
      b1v.x = h[kg * OUT_F + n0 + 16 + mrow];
      b1v.y = h[(kg + 1) * OUT_F + n0 + 16 + mrow];

      c0 = __builtin_amdgcn_wmma_f32_16x16x4_f32(false, pa, false, b0v,
                                                 (short)0, c0, false, false);
      c1 = __builtin_amdgcn_wmma_f32_16x16x4_f32(false, pa, false, b1v,
                                                 (short)0, c1, false, false);
    }
    sp += __shfl_xor(sp, 16, 32);       // both column-halves of this row
    s_run += sp;

    __syncthreads();                    // done reading buf before it refills
  }

  // ---- epilogue: normalize by row sum, ELU, store ----
  const float inv = 1.0f / s_run;       // every row has its diagonal => s>0
  #pragma unroll
  for (int r = 0; r < 8; ++r) {
    const float ir  = __shfl(inv, r + 8 * khalf, 32);
    const int  orow = m0 + r + 8 * khalf;
    const float v0  = c0[r] * ir;
    const float v1  = c1[r] * ir;
    out[orow * OUT_F + n0 + mrow]      = v0 > 0.f ? v0 : __expf(v0) - 1.f;
    out[orow * OUT_F + n0 + 16 + mrow] = v1 > 0.f ? v1 : __expf(v1) - 1.f;
  }
}

// ---------------------------------------------------------------------------
extern "C" void kernel_launch(void* const* d_in, const int* in_sizes, int n_in,
                              void* d_out, int out_size, void* d_ws, size_t ws_size,
                              hipStream_t stream) {
  const float* x   = (const float*)d_in[0];  // (8192, 512)
  const float* adj = (const float*)d_in[1];  // (8192, 8192)
  const float* W   = (const float*)d_in[2];  // (512, 256)
  const float* a   = (const float*)d_in[3];  // (512, 1)
  float* out = (float*)d_out;                // (8192, 256)

  // workspace layout (floats): h | att_l | att_r
  float* h     = (float*)d_ws;
  float* att_l = h + (size_t)N_NODES * OUT_F;
  float* att_r = att_l + N_NODES;

  gat_xw_kernel<<<1024, 256, 0, stream>>>(x, W, h);
  gat_att_kernel<<<1024, 256, 0, stream>>>(h, a, att_l, att_r);
  gat_fused_kernel<<<512, 256, 0, stream>>>(adj, h, att_l, att_r, out);
}